// OPSDLoss_76828374991742
// MI455X (gfx1250) — compile-verified
//
#include <hip/hip_runtime.h>
#include <stdint.h>

#ifndef __has_builtin
#define __has_builtin(x) 0
#endif

#define NTHREADS 256
#define DEPTH 8           // async chunks in flight per wave

#if __has_builtin(__builtin_amdgcn_s_wait_asynccnt)
#define USE_WAIT_BUILTIN 1
#else
#define USE_WAIT_BUILTIN 0
#endif

// ---- CDNA5 async copy: global -> LDS, 16B per lane, tracked by ASYNCcnt ----
__device__ __forceinline__ void async_copy_b128(const float* __restrict__ g,
                                                float4* l) {
  // low 32 bits of the flat address of an LDS object == LDS byte offset
  unsigned lds_off = (unsigned)(uintptr_t)l;
  asm volatile("global_load_async_to_lds_b128 %0, %1, off"
               :: "v"(lds_off), "v"(g)
               : "memory");
}

template <int N>
__device__ __forceinline__ void wait_async() {
#if USE_WAIT_BUILTIN
  __builtin_amdgcn_s_wait_asynccnt(N);
#else
  asm volatile("s_wait_asynccnt %0" :: "n"(N) : "memory");
#endif
}

// One block per (b,t) row: online log-sum-exp over V, then top-K gather + KL.
__global__ __launch_bounds__(NTHREADS) void opsd_row_kernel(
    const float* __restrict__ logits,
    const int*   __restrict__ topk_ids,
    const float* __restrict__ topk_lps,
    const float* __restrict__ mask,
    float* __restrict__ kl_out,
    int V, int K)
{
  __shared__ float4 abuf[DEPTH][NTHREADS];   // 32 KB staging
  __shared__ float wm[NTHREADS / 32], wsv[NTHREADS / 32];

  const int tid = threadIdx.x;
  const int row = blockIdx.x;
  const size_t rowstart = (size_t)row * (size_t)V;
  const float* rp = logits + rowstart;

  // align row to 16B: V odd => row starts rotate mod 16B
  int pre = (int)((4u - ((unsigned)rowstart & 3u)) & 3u);
  if (pre > V) pre = V;
  const int nvec  = (V - pre) >> 2;      // float4 count
  const int nfull = nvec >> 8;           // full 256-float4 (4KB) chunks
  const int vrem  = nvec & 255;
  const int tailc = (V - pre) & 3;
  const float* vb = rp + pre;            // 16B aligned

  float m = -3.0e38f, s = 0.0f;          // online softmax state (finite neutral)

  auto upd1 = [&](float x) {
    float nm = fmaxf(m, x);
    s = s * __expf(m - nm) + __expf(x - nm);
    m = nm;
  };
  auto upd4 = [&](float p0, float p1, float p2, float p3) {
    float cm = fmaxf(fmaxf(p0, p1), fmaxf(p2, p3));
    if (cm > m) { s *= __expf(m - cm); m = cm; }
    s += __expf(p0 - m) + __expf(p1 - m) + __expf(p2 - m) + __expf(p3 - m);
  };
  auto consume = [&](int c) {
    // plain float4 load -> single ds_load_b128; ordering vs the async LDS
    // writes is enforced by the "memory" clobbers on wait_async/async_copy
    float4 v = abuf[c & (DEPTH - 1)][tid];
    upd4(v.x, v.y, v.z, v.w);
  };

  // ---- async-pipelined stream of the row (DEPTH chunks in flight) ----
  const int npro = nfull < DEPTH ? nfull : DEPTH;
  for (int c = 0; c < npro; ++c)
    async_copy_b128(vb + ((size_t)c << 10) + (tid << 2), &abuf[c][tid]);

  int j = 0;
  for (; j + DEPTH < nfull; ++j) {
    async_copy_b128(vb + ((size_t)(j + DEPTH) << 10) + (tid << 2),
                    &abuf[(j + DEPTH) & (DEPTH - 1)][tid]);
    wait_async<DEPTH - 1>();             // oldest chunk landed in LDS
    consume(j);
  }
  wait_async<0>();
  for (; j < nfull; ++j) consume(j);

  // ragged vector remainder + unaligned head + scalar tail (direct loads)
  if (tid < vrem) {
    const float4* q = (const float4*)vb + (((size_t)nfull << 8) + tid);
    float4 v = *q;
    upd4(v.x, v.y, v.z, v.w);
  }
  if (tid < pre)   upd1(rp[tid]);
  if (tid < tailc) upd1(vb[((size_t)nvec << 2) + tid]);

  // ---- wave32 butterfly combine of (m, s) ----
  const int lane = tid & 31, w = tid >> 5;
  for (int o = 16; o; o >>= 1) {
    float om = __shfl_xor(m, o, 32);
    float os = __shfl_xor(s, o, 32);
    float nm = fmaxf(m, om);
    s = s * __expf(m - nm) + os * __expf(om - nm);
    m = nm;
  }
  if (lane == 0) { wm[w] = m; wsv[w] = s; }
  __syncthreads();

  if (tid < 32) {
    const int nw = NTHREADS / 32;
    m = (lane < nw) ? wm[lane] : -3.0e38f;
    s = (lane < nw) ? wsv[lane] : 0.0f;
    for (int o = 16; o; o >>= 1) {
      float om = __shfl_xor(m, o, 32);
      float os = __shfl_xor(s, o, 32);
      float nm = fmaxf(m, om);
      s = s * __expf(m - nm) + os * __expf(om - nm);
      m = nm;
    }
    // butterfly leaves identical (M,S) in every lane
    const float logZ = m + __logf(s);

    // ---- top-K gather + renormalized-teacher KL (lanes 0..K-1) ----
    float wgt = 0.0f, tlp = 0.0f, x = 0.0f;
    if (lane < K) {
      int id = topk_ids[(size_t)row * K + lane];
      x   = logits[rowstart + (size_t)id];
      tlp = topk_lps[(size_t)row * K + lane];
      wgt = __expf(tlp);
    }
    float denom = wgt;
    for (int o = 16; o; o >>= 1) denom += __shfl_xor(denom, o, 32);
    denom = fmaxf(denom, 1e-8f);
    const float logd = __logf(denom);

    float klk = 0.0f;
    if (lane < K) {
      float tp  = wgt / denom;
      float slp = x - logZ;              // student log-prob at teacher id
      klk = tp * ((tlp - logd) - slp);   // tp * (log tp - student_lp)
    }
    for (int o = 16; o; o >>= 1) klk += __shfl_xor(klk, o, 32);
    if (lane == 0) kl_out[row] = klk * mask[row];
  }
}

// Tiny finalize: per-response masked mean, then mean over batch.
__global__ __launch_bounds__(256) void opsd_finalize_kernel(
    const float* __restrict__ klr, const float* __restrict__ mask,
    float* __restrict__ out, int B, int T)
{
  __shared__ float red[32];
  const int tid = threadIdx.x;
  const int lane = tid & 31, w = tid >> 5;
  const int nw = blockDim.x >> 5;
  float acc = 0.0f;
  for (int b = 0; b < B; ++b) {
    float skl = 0.0f, sm = 0.0f;
    for (int t = tid; t < T; t += blockDim.x) {
      skl += klr[(size_t)b * T + t];
      sm  += mask[(size_t)b * T + t];
    }
    for (int o = 16; o; o >>= 1) {
      skl += __shfl_xor(skl, o, 32);
      sm  += __shfl_xor(sm, o, 32);
    }
    if (lane == 0) { red[w] = skl; red[w + 16] = sm; }
    __syncthreads();
    if (tid == 0) {
      float tk = 0.0f, tm = 0.0f;
      for (int i = 0; i < nw; ++i) { tk += red[i]; tm += red[i + 16]; }
      acc += tk / fmaxf(tm, 1.0f);
    }
    __syncthreads();
  }
  if (tid == 0) out[0] = acc / (float)B;
}

extern "C" void kernel_launch(void* const* d_in, const int* in_sizes, int n_in,
                              void* d_out, int out_size, void* d_ws, size_t ws_size,
                              hipStream_t stream) {
  (void)n_in; (void)out_size; (void)ws_size;
  const float* logits   = (const float*)d_in[0];
  // d_in[1] = completion_ids (unused by the reference math)
  const int*   topk_ids = (const int*)d_in[2];
  const float* topk_lps = (const float*)d_in[3];
  const float* mask     = (const float*)d_in[4];
  // d_in[5] = prompt_lens (unused by the reference math)

  const int BT = in_sizes[4];                    // B*T
  const int B  = in_sizes[5];                    // prompt_lens length
  const int T  = BT / B;
  const int K  = in_sizes[3] / BT;
  const int V  = (int)((long long)in_sizes[0] / (long long)BT);

  float* klr = (float*)d_ws;                     // BT floats of scratch

  opsd_row_kernel<<<BT, NTHREADS, 0, stream>>>(logits, topk_ids, topk_lps,
                                               mask, klr, V, K);
  opsd_finalize_kernel<<<1, 256, 0, stream>>>(klr, mask, (float*)d_out, B, T);
}